// Pooler_15513421873663
// MI455X (gfx1250) — compile-verified
//
#include <hip/hip_runtime.h>
#include <math.h>

typedef __attribute__((ext_vector_type(2))) float v2f;
typedef __attribute__((ext_vector_type(8))) float v8f;

#define OUTD 7
#define SP 14            // OUT*SR sample grid
#define CG 32            // channels per block
#define CTOT 256
#define MROWS (CG * SP)  // 448 = stage-X M  (m = r*32 + ch)
#define NCOLS (CG * OUTD)// 224 = stage-Y N  (n = ch*7 + ox)
#define LDV 18           // padded LDS K-stride (floats): even (b64 align) + bank spread
#define NWAVE 7
#define BLK (NWAVE * 32) // 224 threads

__global__ __launch_bounds__(BLK) void roi_pool_wmma_kernel(
    const float* __restrict__ f0, const float* __restrict__ f1,
    const float* __restrict__ f2, const float* __restrict__ f3,
    const float* __restrict__ boxes, const int* __restrict__ imgids,
    float* __restrict__ out)
{
    // A staging for stage X: val[(r*32 + ch)][sx], K padded to 16 (cols 14,15 zero)
    __shared__ __align__(16) float lds_val[MROWS * LDV];
    // B staging for stage Y: U[(ch*7 + ox)][r], K padded to 16 (cols 14,15 zero)
    __shared__ __align__(16) float lds_u[NCOLS * LDV];

    const int t   = threadIdx.x;
    const int box = blockIdx.x;
    const int cg  = blockIdx.y;      // channel group 0..7

    // ---- per-box params (uniform across block; scalar-cached loads) ----
    const float bx1 = boxes[box * 4 + 0];
    const float by1 = boxes[box * 4 + 1];
    const float bx2 = boxes[box * 4 + 2];
    const float by2 = boxes[box * 4 + 3];
    const int   img = imgids[box];

    const float area = (bx2 - bx1) * (by2 - by1);
    const float srt  = sqrtf(area);
    float lv = floorf(4.0f + log2f(srt * (1.0f / 224.0f) + 1e-6f));
    lv = fminf(fmaxf(lv, 2.0f), 5.0f);
    const int li = (int)lv - 2;

    const float* F; int H, W; float scale;
    if (li == 0)      { F = f0; H = 200; W = 200; scale = 0.25f;    }
    else if (li == 1) { F = f1; H = 100; W = 100; scale = 0.125f;   }
    else if (li == 2) { F = f2; H = 50;  W = 50;  scale = 0.0625f;  }
    else              { F = f3; H = 25;  W = 25;  scale = 0.03125f; }
    const int HW = H * W;

    // ---- zero K-pad columns (sx/r = 14,15) ----
    for (int i = t; i < MROWS * 2; i += BLK)
        lds_val[(i >> 1) * LDV + 14 + (i & 1)] = 0.0f;
    for (int i = t; i < NCOLS * 2; i += BLK)
        lds_u[(i >> 1) * LDV + 14 + (i & 1)] = 0.0f;

    // ---- Phase 1: gather + bilinear. One sample point per thread (196 of 224),
    //      weights computed ONCE and reused across all 32 channels. ----
    if (t < SP * SP) {
        const int r  = t / SP;   // sample row  (oy*2+sy)
        const int sx = t % SP;   // sample col  (ox*2+sx)

        const float x1s = bx1 * scale, y1s = by1 * scale;
        const float roi_w = fmaxf(bx2 * scale - x1s, 1.0f);
        const float roi_h = fmaxf(by2 * scale - y1s, 1.0f);
        const float bw = roi_w * (1.0f / OUTD);
        const float bh = roi_h * (1.0f / OUTD);

        const float y = y1s + (float)(r  >> 1) * bh + ((float)(r  & 1) + 0.5f) * bh * 0.5f;
        const float x = x1s + (float)(sx >> 1) * bw + ((float)(sx & 1) + 0.5f) * bw * 0.5f;
        const bool valid = (y >= -1.0f) && (y <= (float)H) && (x >= -1.0f) && (x <= (float)W);

        const float yc = fminf(fmaxf(y, 0.0f), (float)(H - 1));
        const float xc = fminf(fmaxf(x, 0.0f), (float)(W - 1));
        const int y0 = (int)floorf(yc);
        const int x0 = (int)floorf(xc);
        const int y1i = min(y0 + 1, H - 1);
        const int x1i = min(x0 + 1, W - 1);
        const float ly = yc - (float)y0, lx = xc - (float)x0;
        const float hy = 1.0f - ly, hx = 1.0f - lx;
        float w00 = hy * hx, w01 = hy * lx, w10 = ly * hx, w11 = ly * lx;
        if (!valid) { w00 = w01 = w10 = w11 = 0.0f; }

        const int o00 = y0 * W + x0,  o01 = y0 * W + x1i;
        const int o10 = y1i * W + x0, o11 = y1i * W + x1i;

        const float* plane0 = F + ((size_t)img * CTOT + (size_t)cg * CG) * (size_t)HW;
        float* dst = &lds_val[(r * CG) * LDV + sx];   // + ch*LDV per channel (m = r*32+ch)

        #pragma unroll 8
        for (int ch = 0; ch < CG; ++ch) {
            const float* p = plane0 + (size_t)ch * HW;
            const float v = w00 * p[o00] + w01 * p[o01] + w10 * p[o10] + w11 * p[o11];
            dst[ch * LDV] = v;
        }
    }
    __syncthreads();

    const int lane   = t & 31;
    const int wv     = t >> 5;          // wave id 0..6
    const int lhalf  = (lane >> 4);     // 0: K=0,1  1: K=2,3 (A/B fragment halves)
    const int l16    = lane & 15;

    // ---- Phase 2 (stage X): U = val(448x14) * Ax(14x7), fp32 WMMA 16x16x4.
    //      4 M-tiles per wave; Ax synthesized in registers. ----
    for (int mt = wv * 4; mt < wv * 4 + 4; ++mt) {
        v8f acc = {0.f, 0.f, 0.f, 0.f, 0.f, 0.f, 0.f, 0.f};
        #pragma unroll
        for (int kk = 0; kk < 4; ++kk) {
            const int sxb = kk * 4 + lhalf * 2;           // K index of VGPR0 in this lane
            const int m   = mt * 16 + l16;                // A row (both lane halves)
            const v2f a   = *(const v2f*)&lds_val[m * LDV + sxb];
            const float b0 = (sxb     < 14 && ((sxb)     >> 1) == l16) ? 0.5f : 0.0f;
            const float b1 = (sxb + 1 < 14 && ((sxb + 1) >> 1) == l16) ? 0.5f : 0.0f;
            const v2f b = {b0, b1};
            acc = __builtin_amdgcn_wmma_f32_16x16x4_f32(
                false, a, false, b, (short)0, acc, false, false);
        }
        // D layout: lane holds column n=l16 (=ox), rows v + lhalf*8
        const int ox = l16;
        if (ox < OUTD) {
            #pragma unroll
            for (int v = 0; v < 8; ++v) {
                const int m  = mt * 16 + v + lhalf * 8;
                const int ch = m & (CG - 1);              // m = r*32 + ch: pure bit ops
                const int r  = m >> 5;
                lds_u[(ch * OUTD + ox) * LDV + r] = acc[v];
            }
        }
    }
    __syncthreads();

    // ---- Phase 3 (stage Y): out = Ay(7x14) * U(14x224), fp32 WMMA 16x16x4.
    //      2 N-tiles per wave; Ay synthesized in registers; D written to global.
    //      n = ch*7+ox keeps output stores as contiguous 7-float runs per channel. ----
    for (int nt = wv * 2; nt < wv * 2 + 2; ++nt) {
        v8f acc = {0.f, 0.f, 0.f, 0.f, 0.f, 0.f, 0.f, 0.f};
        const int n = nt * 16 + l16;
        #pragma unroll
        for (int kk = 0; kk < 4; ++kk) {
            const int rb = kk * 4 + lhalf * 2;            // K index of VGPR0
            const v2f b  = *(const v2f*)&lds_u[n * LDV + rb];
            const int oy = l16;                           // A row (both halves)
            const float a0 = (oy < OUTD && rb     < 14 && ((rb)     >> 1) == oy) ? 0.5f : 0.0f;
            const float a1 = (oy < OUTD && rb + 1 < 14 && ((rb + 1) >> 1) == oy) ? 0.5f : 0.0f;
            const v2f a = {a0, a1};
            acc = __builtin_amdgcn_wmma_f32_16x16x4_f32(
                false, a, false, b, (short)0, acc, false, false);
        }
        const int ch = n / OUTD, ox = n % OUTD;
        float* obase = out + ((size_t)box * CTOT + (size_t)cg * CG + ch) * (OUTD * OUTD) + ox;
        if (lhalf == 0) {   // D rows (oy) 0..7 live in lanes 0-15; oy<7 valid
            #pragma unroll
            for (int v = 0; v < OUTD; ++v)
                obase[v * OUTD] = acc[v];
        }
    }
}

extern "C" void kernel_launch(void* const* d_in, const int* in_sizes, int n_in,
                              void* d_out, int out_size, void* d_ws, size_t ws_size,
                              hipStream_t stream) {
    const float* f0    = (const float*)d_in[0];
    const float* f1    = (const float*)d_in[1];
    const float* f2    = (const float*)d_in[2];
    const float* f3    = (const float*)d_in[3];
    const float* boxes = (const float*)d_in[4];
    const int*   ids   = (const int*)d_in[5];
    float*       out   = (float*)d_out;

    const int nBoxes = in_sizes[4] / 4;   // boxes are (N,4)
    if (nBoxes <= 0) return;

    dim3 grid((unsigned)nBoxes, CTOT / CG);   // (1000, 8)
    roi_pool_wmma_kernel<<<grid, BLK, 0, stream>>>(f0, f1, f2, f3, boxes, ids, out);
}